// SpatialTransformer_75617194213396
// MI455X (gfx1250) — compile-verified
//
#include <hip/hip_runtime.h>
#include <hip/hip_bf16.h>
#include <math.h>

typedef __attribute__((ext_vector_type(2))) float v2f;
typedef __attribute__((ext_vector_type(8))) float v8f;

#define B_      64
#define H_      128
#define W_      2048
#define HW_     (H_ * W_)         // 262144
#define OUT_H_  128
#define OUT_W_  2048

#define NBLK1   256               // K-partition blocks for GEMM phase
#define KTOT    (HW_ / NBLK1)     // 1024 K per block
#define KCHUNK  128               // K staged in LDS per iteration
#define KITER   (KTOT / KCHUNK)   // 8
#define SP      (KCHUNK + 4)      // padded LDS row stride (floats) -> conflict-free

#define PART_FLOATS (NBLK1 * 8 * 256)   // 524288 floats = 2 MB of partial C tiles

// -------------------------------------------------------------------------
// Kernel 1: partial GEMM  theta_part = x(64 x K) @ wloc(K x 16padded)
// using V_WMMA_F32_16X16X4_F32.  8 waves/block: wave w -> M-tile (w&3),
// K-half (w>>2).  A/B fragments staged via LDS for coalesced global traffic.
// -------------------------------------------------------------------------
__global__ void __launch_bounds__(256)
k1_gemm_partial(const float* __restrict__ x, const float* __restrict__ wloc,
                float* __restrict__ part) {
    __shared__ float xs_[B_ * SP];          // 64 x 132 floats = 33792 B
    __shared__ float wsh[KCHUNK * 16];      // 128 x 16 floats =  8192 B

    const int tid   = threadIdx.x;
    const int lane  = tid & 31;
    const int wv    = tid >> 5;     // 0..7
    const int mtile = wv & 3;       // 16-batch tile
    const int kh    = wv >> 2;      // 0/1 -> K half of the chunk
    const int m16   = lane & 15;    // A: row M, B: col N
    const int lkh   = lane >> 4;    // lane half selects K pair {0,1} vs {2,3}
    const long kbase0 = (long)blockIdx.x * KTOT;

    // zero-fill w tile once; columns 6..15 stay zero forever (padding)
    for (int i = tid; i < KCHUNK * 16; i += 256) wsh[i] = 0.0f;

    v8f c = {};   // f32 accumulator fragment, persists across K chunks

    for (int it = 0; it < KITER; ++it) {
        const long kbase = kbase0 + (long)it * KCHUNK;
        __syncthreads();    // LDS safe to overwrite (also orders the zero-fill)

        // ---- stage x tile: 64 rows x 128 floats, b128 coalesced loads ----
        {
            const int c4 = (tid & 31) * 4;   // float4 column
            const int mr = tid >> 5;         // 8 rows in flight
#pragma unroll
            for (int i = 0; i < 8; ++i) {
                const int m = mr + 8 * i;
                const float4 v = *(const float4*)(x + (long)m * HW_ + kbase + c4);
                *(float4*)(&xs_[m * SP + c4]) = v;
            }
        }
        // ---- stage w tile: 128 x 6 floats -> [k][n(16)] layout ----
#pragma unroll
        for (int i = 0; i < 3; ++i) {
            const int j = tid + i * 256;     // 0..767
            const int kk = j / 6, n = j % 6;
            wsh[kk * 16 + n] = wloc[kbase * 6 + j];
        }
        __syncthreads();

        // ---- WMMA over this chunk's K half (64 k -> 16 wmma ops) ----
        const float* arow = &xs_[(mtile * 16 + m16) * SP + 2 * lkh];
#pragma unroll
        for (int kk = kh * 64; kk < kh * 64 + 64; kk += 4) {
            v2f a = *(const v2f*)(arow + kk);               // A: [m16, kk+2*lkh .. +1]
            v2f b;
            b.x = wsh[(kk + 2 * lkh + 0) * 16 + m16];       // B: [k, n=m16]
            b.y = wsh[(kk + 2 * lkh + 1) * 16 + m16];
            c = __builtin_amdgcn_wmma_f32_16x16x4_f32(
                    false, a, false, b, (short)0, c, false, false);
        }
    }

    // ---- write 16x16 C tile row-major to workspace ----
    float* dst = part + ((long)blockIdx.x * 8 + wv) * 256;
#pragma unroll
    for (int r = 0; r < 8; ++r)
        dst[(r + 8 * lkh) * 16 + m16] = c[r];
}

// -------------------------------------------------------------------------
// Kernel 2: deterministic reduction of 512 partial tiles -> theta (64 x 6)
// blockIdx.x = batch m; 256 threads: n = t>>4, sub-reducer = t&15
// -------------------------------------------------------------------------
__global__ void __launch_bounds__(256)
k2_reduce(const float* __restrict__ part, const float* __restrict__ bloc,
          float* __restrict__ theta) {
    __shared__ float red[16][17];
    const int m = blockIdx.x;
    const int mtile = m >> 4, mi = m & 15;
    const int tid = threadIdx.x;
    const int n = tid >> 4, sub = tid & 15;

    float s = 0.0f;
#pragma unroll 4
    for (int i = 0; i < 32; ++i) {
        const int idx = sub * 32 + i;        // 0..511, fixed order
        const int blk = idx >> 1, khalf = idx & 1;
        s += part[((blk * 8) + khalf * 4 + mtile) * 256 + mi * 16 + n];
    }
    red[n][sub] = s;
    __syncthreads();
    if (sub == 0 && n < 6) {
        float t = 0.0f;
#pragma unroll
        for (int i = 0; i < 16; ++i) t += red[n][i];
        theta[m * 6 + n] = t + bloc[n];
    }
}

// -------------------------------------------------------------------------
// Kernel 3: affine bilinear sampling. 1 px/thread; theta ~ identity so the
// four gathers are near-coalesced and hit L2 (x fits in the 192 MB L2).
// -------------------------------------------------------------------------
__global__ void __launch_bounds__(256)
k3_sample(const float* __restrict__ x, const float* __restrict__ theta,
          float* __restrict__ out) {
    const int b  = blockIdx.z;
    const int oy = blockIdx.y;
    const int ox = blockIdx.x * 256 + threadIdx.x;

    const float t0 = theta[b * 6 + 0], t1 = theta[b * 6 + 1], t2 = theta[b * 6 + 2];
    const float t3 = theta[b * 6 + 3], t4 = theta[b * 6 + 4], t5 = theta[b * 6 + 5];

    const float fx = (float)ox, fy = (float)oy;
    const float xq = t0 * fx + t1 * fy + t2;
    const float yq = t3 * fx + t4 * fy + t5;

    const int x0 = (int)floorf(xq);
    const int y0 = (int)floorf(yq);
    const int x0c = min(max(x0, 0), W_ - 1);
    const int x1c = min(max(x0 + 1, 0), W_ - 1);
    const int y0c = min(max(y0, 0), H_ - 1);
    const int y1c = min(max(y0 + 1, 0), H_ - 1);

    const float* xb = x + (long)b * HW_;
    const float Ia = xb[y0c * W_ + x0c];
    const float Ib = xb[y1c * W_ + x0c];
    const float Ic = xb[y0c * W_ + x1c];
    const float Id = xb[y1c * W_ + x1c];

    const float x0f = (float)x0c, x1f = (float)x1c;
    const float y0f = (float)y0c, y1f = (float)y1c;
    const float wa = (x1f - xq) * (y1f - yq);
    const float wb = (x1f - xq) * (yq - y0f);
    const float wc = (xq - x0f) * (y1f - yq);
    const float wd = (xq - x0f) * (yq - y0f);

    const float r = wa * Ia + wb * Ib + wc * Ic + wd * Id;
    __builtin_nontemporal_store(r, out + ((long)b * OUT_H_ + oy) * OUT_W_ + ox);
}

// -------------------------------------------------------------------------
extern "C" void kernel_launch(void* const* d_in, const int* in_sizes, int n_in,
                              void* d_out, int out_size, void* d_ws, size_t ws_size,
                              hipStream_t stream) {
    const float* x    = (const float*)d_in[0];   // (64,128,2048) f32
    const float* wloc = (const float*)d_in[1];   // (262144,6)    f32
    const float* bloc = (const float*)d_in[2];   // (6,)          f32
    float* out   = (float*)d_out;
    float* part  = (float*)d_ws;                 // 2 MB partial tiles
    float* theta = part + PART_FLOATS;           // 384 floats

    k1_gemm_partial<<<NBLK1, 256, 0, stream>>>(x, wloc, part);
    k2_reduce<<<B_, 256, 0, stream>>>(part, bloc, theta);
    k3_sample<<<dim3(OUT_W_ / 256, OUT_H_, B_), 256, 0, stream>>>(x, theta, out);
}